// ImgtoClass_Metric_69647189672573
// MI455X (gfx1250) — compile-verified
//
#include <hip/hip_runtime.h>
#include <math.h>

typedef float v2f __attribute__((ext_vector_type(2)));
typedef float v8f __attribute__((ext_vector_type(8)));

#define WIN   11
#define HW    21
#define PDIM  32       // padded plane dim
#define PPIX  1024     // 32*32 padded plane
#define NPIX  441      // 21*21
#define OUTP  11       // 21-11+1
#define OPIX  121      // 11*11
#define NIMG  40       // 15 queries + 25 supports
#define NCH   64
#define SSIM_C1 1.0e-4f
#define SSIM_C2 9.0e-4f

// ---------------------------------------------------------------------------
// V_WMMA_F32_16X16X4_F32: D = A(16x4 f32) * B(4x16 f32) + C(16x16 f32)
// ---------------------------------------------------------------------------
__device__ __forceinline__ v8f wmma4(v2f a, v2f b, v8f c) {
  return __builtin_amdgcn_wmma_f32_16x16x4_f32(
      /*neg_a=*/false, a, /*neg_b=*/false, b,
      /*c_mod=*/(short)0, c, /*reuse_a=*/false, /*reuse_b=*/false);
}

// Build Gaussian window + band matrix A (32x16, A[r][j] = win[r-j]) in LDS,
// then hoist this thread's per-kstep (k,k+1) pairs into registers. These
// pairs serve as the pass-1 B operand AND the pass-2 A^T operand.
__device__ __forceinline__ void build_gauss(float* win_s, float* Aband,
                                            v2f* bband, int m, int hi) {
  const int tid = threadIdx.x;
  if (tid == 0) {
    float g[WIN];
    float ssum = 0.f;
#pragma unroll
    for (int i = 0; i < WIN; ++i) {
      float d = (float)i - 5.0f;
      g[i] = expf(-(d * d) / 4.5f);
      ssum += g[i];
    }
#pragma unroll
    for (int i = 0; i < WIN; ++i) win_s[i] = g[i] / ssum;
  }
  __syncthreads();
  for (int idx = tid; idx < 512; idx += blockDim.x) {
    const int r = idx >> 4, j = idx & 15;
    const int k = r - j;
    Aband[idx] = (j < OUTP && r < HW && k >= 0 && k < WIN) ? win_s[k] : 0.f;
  }
  __syncthreads();
#pragma unroll
  for (int ks = 0; ks < 8; ++ks) {
    const int k = ks * 4 + 2 * hi;
    bband[ks].x = Aband[k * 16 + m];
    bband[ks].y = Aband[(k + 1) * 16 + m];
  }
}

// Separable blur of a padded 32x32 plane: Z = A^T * (P * A), all via WMMA.
// fetch2(r,k) returns v2f {P[r][k], P[r][k+1]} (unconditional, pad is zero).
// Result element (i = v + 8*hi, j = lane&15).
template <typename Fetch2T>
__device__ __forceinline__ v8f blur_wmma(Fetch2T fetch2, const v2f* bband,
                                         float* yhw, int m, int hi) {
#pragma unroll
  for (int mt = 0; mt < 2; ++mt) {
    v8f acc = {0.f, 0.f, 0.f, 0.f, 0.f, 0.f, 0.f, 0.f};
#pragma unroll
    for (int ks = 0; ks < 8; ++ks) {
      const int k = ks * 4 + 2 * hi;
      acc = wmma4(fetch2(mt * 16 + m, k), bband[ks], acc);
    }
#pragma unroll
    for (int v = 0; v < 8; ++v)
      yhw[(mt * 16 + v + 8 * hi) * 16 + m] = acc[v];
  }
  asm volatile("s_wait_dscnt 0x0" ::: "memory");  // wave-local LDS RAW fence
  v8f acc = {0.f, 0.f, 0.f, 0.f, 0.f, 0.f, 0.f, 0.f};
#pragma unroll
  for (int ks = 0; ks < 8; ++ks) {
    const int k = ks * 4 + 2 * hi;
    v2f bb;
    bb.x = yhw[k * 16 + m];
    bb.y = yhw[(k + 1) * 16 + m];
    acc = wmma4(bband[ks], bb, acc);
  }
  return acc;
}

// Fused dual blur: z0 = blur(P), z1 = blur(P∘P); plane fetched once.
__device__ __forceinline__ void blur2_wmma(const float* __restrict__ plane,
                                           const v2f* bband, float* y0,
                                           float* y1, int m, int hi, v8f& z0,
                                           v8f& z1) {
#pragma unroll
  for (int mt = 0; mt < 2; ++mt) {
    v8f a0 = {0.f, 0.f, 0.f, 0.f, 0.f, 0.f, 0.f, 0.f};
    v8f a1 = {0.f, 0.f, 0.f, 0.f, 0.f, 0.f, 0.f, 0.f};
#pragma unroll
    for (int ks = 0; ks < 8; ++ks) {
      const int k = ks * 4 + 2 * hi;
      const v2f a  = *(const v2f*)(plane + (mt * 16 + m) * PDIM + k);
      const v2f a2 = a * a;
      a0 = wmma4(a, bband[ks], a0);
      a1 = wmma4(a2, bband[ks], a1);
    }
#pragma unroll
    for (int v = 0; v < 8; ++v) {
      y0[(mt * 16 + v + 8 * hi) * 16 + m] = a0[v];
      y1[(mt * 16 + v + 8 * hi) * 16 + m] = a1[v];
    }
  }
  asm volatile("s_wait_dscnt 0x0" ::: "memory");
  v8f a0 = {0.f, 0.f, 0.f, 0.f, 0.f, 0.f, 0.f, 0.f};
  v8f a1 = {0.f, 0.f, 0.f, 0.f, 0.f, 0.f, 0.f, 0.f};
#pragma unroll
  for (int ks = 0; ks < 8; ++ks) {
    const int k = ks * 4 + 2 * hi;
    v2f b0, b1;
    b0.x = y0[k * 16 + m];
    b0.y = y0[(k + 1) * 16 + m];
    b1.x = y1[k * 16 + m];
    b1.y = y1[(k + 1) * 16 + m];
    a0 = wmma4(bband[ks], b0, a0);
    a1 = wmma4(bband[ks], b1, a1);
  }
  z0 = a0;
  z1 = a1;
}

// ---------------------------------------------------------------------------
// Kernel 1: channel-L2 normalization into zero-padded 32x32 planes.
// One wave per (img, padded pixel); lanes cover the 64 contiguous channels.
// Pad positions (r>=21 || c>=21) are written as real zeros every launch.
// ---------------------------------------------------------------------------
__global__ void __launch_bounds__(256)
norm_kernel(const float* __restrict__ x1, const float* __restrict__ x2,
            float* __restrict__ wsNorm) {
  const int gw   = (int)((blockIdx.x * blockDim.x + threadIdx.x) >> 5);
  const int lane = threadIdx.x & 31;
  if (gw >= NIMG * PPIX) return;
  const int img  = gw >> 10;
  const int ppad = gw & (PPIX - 1);
  const int r = ppad >> 5, c = ppad & 31;
  float* dst = wsNorm + (size_t)img * NCH * PPIX;
  if (r < HW && c < HW) {  // uniform across the wave
    const int p = r * HW + c;
    const float* src = (img < 15)
                           ? (x1 + ((size_t)img * NPIX + p) * NCH)
                           : (x2 + ((size_t)(img - 15) * NPIX + p) * NCH);
    const float v0 = src[lane];
    const float v1 = src[lane + 32];
    float ss = v0 * v0 + v1 * v1;
#pragma unroll
    for (int off = 16; off > 0; off >>= 1) ss += __shfl_xor(ss, off, 32);
    const float inv = 1.0f / sqrtf(ss);
    dst[(size_t)lane * PPIX + ppad]        = v0 * inv;
    dst[(size_t)(lane + 32) * PPIX + ppad] = v1 * inv;
  } else {
    dst[(size_t)lane * PPIX + ppad]        = 0.f;
    dst[(size_t)(lane + 32) * PPIX + ppad] = 0.f;
  }
}

// ---------------------------------------------------------------------------
// Kernel 2: per-image per-channel stats: mu = blur(v), msq = blur(v^2),
// fused so the plane is fetched once. One block per image, 8 ch per wave.
// ---------------------------------------------------------------------------
__global__ void __launch_bounds__(256)
stats_kernel(const float* __restrict__ wsNorm, float* __restrict__ wsMu,
             float* __restrict__ wsMsq) {
  __shared__ float win_s[WIN];
  __shared__ float Aband[512];
  __shared__ float yh[8][2][512];
  const int tid  = threadIdx.x;
  const int lane = tid & 31;
  const int wave = tid >> 5;
  const int m    = lane & 15;
  const int hi   = lane >> 4;
  const int img  = blockIdx.x;
  v2f bband[8];
  build_gauss(win_s, Aband, bband, m, hi);

  for (int cc = 0; cc < 8; ++cc) {
    const int c = wave + 8 * cc;
    const float* plane = wsNorm + ((size_t)img * NCH + c) * PPIX;
    v8f mu, ms;
    blur2_wmma(plane, bband, yh[wave][0], yh[wave][1], m, hi, mu, ms);
    float* muo = wsMu  + ((size_t)img * NCH + c) * OPIX;
    float* mso = wsMsq + ((size_t)img * NCH + c) * OPIX;
#pragma unroll
    for (int v = 0; v < 8; ++v) {
      const int i = v + 8 * hi;
      if (i < OUTP && m < OUTP) {
        muo[i * OUTP + m] = mu[v];
        mso[i * OUTP + m] = ms[v];
      }
    }
  }
}

// ---------------------------------------------------------------------------
// Kernel 3: one block per (query b, class n); 8 waves share 5 shots x 64
// channels. blur(x∘y) via WMMA, branch-free SSIM combine (mask by cndmask),
// deterministic fixed-order reduction. out[b*5+n] = sum_s mean(ssim_map).
// ---------------------------------------------------------------------------
__global__ void __launch_bounds__(256)
pair_kernel(const float* __restrict__ wsNorm, const float* __restrict__ wsMu,
            const float* __restrict__ wsMsq, float* __restrict__ out) {
  __shared__ float win_s[WIN];
  __shared__ float Aband[512];
  __shared__ float yh[8][512];
  __shared__ float wsum[8];
  const int tid  = threadIdx.x;
  const int lane = tid & 31;
  const int wave = tid >> 5;
  const int m    = lane & 15;
  const int hi   = lane >> 4;
  const int b    = blockIdx.x / 5;
  const int n    = blockIdx.x % 5;
  v2f bband[8];
  build_gauss(win_s, Aband, bband, m, hi);
  float* yhw = yh[wave];
  const bool mok = (m < OUTP);

  float lsum = 0.f;
  for (int t = wave; t < 5 * NCH; t += 8) {
    const int s = t >> 6;
    const int c = t & 63;
    const int imgs = 15 + n * 5 + s;
    const float* px = wsNorm + ((size_t)b * NCH + c) * PPIX;
    const float* py = wsNorm + ((size_t)imgs * NCH + c) * PPIX;
    auto fprod2 = [&](int r, int k) {
      const v2f xv = *(const v2f*)(px + r * PDIM + k);
      const v2f yv = *(const v2f*)(py + r * PDIM + k);
      return xv * yv;
    };
    v8f z = blur_wmma(fprod2, bband, yhw, m, hi);  // blur(x*y)

    const float* muq = wsMu  + ((size_t)b * NCH + c) * OPIX;
    const float* mus = wsMu  + ((size_t)imgs * NCH + c) * OPIX;
    const float* sqq = wsMsq + ((size_t)b * NCH + c) * OPIX;
    const float* sqs = wsMsq + ((size_t)imgs * NCH + c) * OPIX;
#pragma unroll
    for (int v = 0; v < 8; ++v) {
      const int i = v + 8 * hi;
      const int o = i * OUTP + m;  // may index slack region when invalid
      const float mu1 = muq[o];
      const float mu2 = mus[o];
      const float m12 = mu1 * mu2;
      const float s1  = sqq[o] - mu1 * mu1;
      const float s2  = sqs[o] - mu2 * mu2;
      const float s12 = z[v] - m12;
      const float num = (2.f * m12 + SSIM_C1) * (2.f * s12 + SSIM_C2);
      const float den =
          (mu1 * mu1 + mu2 * mu2 + SSIM_C1) * (s1 + s2 + SSIM_C2);
      const float term = num / den;
      lsum += (mok && i < OUTP) ? term : 0.f;  // cndmask, no EXEC branch
    }
  }
  // deterministic reduction: shfl within wave, fixed-order sum across waves
#pragma unroll
  for (int off = 16; off > 0; off >>= 1) lsum += __shfl_xor(lsum, off, 32);
  if (lane == 0) wsum[wave] = lsum;
  __syncthreads();
  if (tid == 0) {
    float tot = 0.f;
    for (int w = 0; w < 8; ++w) tot += wsum[w];
    out[b * 5 + n] = tot / (float)(NCH * OPIX);
  }
}

// ---------------------------------------------------------------------------
extern "C" void kernel_launch(void* const* d_in, const int* in_sizes, int n_in,
                              void* d_out, int out_size, void* d_ws,
                              size_t ws_size, hipStream_t stream) {
  const float* x1 = (const float*)d_in[0];  // (15, 441, 64)
  const float* x2 = (const float*)d_in[1];  // (5, 5, 441, 64)
  float* wsf    = (float*)d_ws;
  float* wsNorm = wsf;  // 40*64*1024 floats, zero-padded 32x32 planes
  float* wsMu   = wsNorm + (size_t)NIMG * NCH * PPIX;   // 40*64*121 (+256 slack)
  float* wsMsq  = wsMu + (size_t)NIMG * NCH * OPIX + 256;  // same (+256 slack)

  const int nWaves = NIMG * PPIX;  // 40960 (img, padded pixel) tasks
  norm_kernel<<<(nWaves + 7) / 8, 256, 0, stream>>>(x1, x2, wsNorm);
  stats_kernel<<<NIMG, 256, 0, stream>>>(wsNorm, wsMu, wsMsq);
  pair_kernel<<<75, 256, 0, stream>>>(wsNorm, wsMu, wsMsq, (float*)d_out);
}